// Simple_Decoder_34608846471829
// MI455X (gfx1250) — compile-verified
//
#include <hip/hip_runtime.h>
#include <math.h>

#define B_ 256
#define L_ 50
#define H_ 1024
#define E_ 512
#define V_ 32000

typedef __attribute__((ext_vector_type(2))) float v2f;
typedef __attribute__((ext_vector_type(8))) float v8f;

// D = A(16x4 f32) x B(4x16 f32) + C, full f32 precision (CDNA5 WMMA).
__device__ __forceinline__ v8f wmma_f32(v2f a, v2f b, v8f c) {
    return __builtin_amdgcn_wmma_f32_16x16x4_f32(
        /*neg_a=*/false, a, /*neg_b=*/false, b,
        /*c_mod=*/(short)0, c, /*reuse_a=*/false, /*reuse_b=*/false);
}

// ---------------------------------------------------------------------------
// Generic GEMM: C[M,N] = A[M,K] @ W[N,K]^T (+ bias[N])
// grid = (N/256, M/32), block = 256 = 8 waves; each wave owns a 32x32 tile
// (2 M-subtiles x 2 N-subtiles -> 4 WMMAs per 4 loads = 1 load/WMMA).
// A frag: lane lp=lane&15 -> row, contiguous K pair at k+2*(lane>>4).
// B frag: lane lp -> W row n, same K pair (B[k][n] = W[n][k]).
// ---------------------------------------------------------------------------
__global__ __launch_bounds__(256) void gemm_xwT(
    const float* __restrict__ A, int lda,
    const float* __restrict__ W, int ldw,
    const float* __restrict__ bias,
    float* __restrict__ C, int ldc, int K)
{
    const int lane = threadIdx.x & 31;
    const int wave = threadIdx.x >> 5;
    const int hi   = lane >> 4;
    const int lp   = lane & 15;
    const int n0 = blockIdx.x * 256 + wave * 32;
    const int m0 = blockIdx.y * 32;

    const float* Ar0 = A + (size_t)(m0 + lp) * lda + 2 * hi;
    const float* Ar1 = Ar0 + (size_t)16 * lda;
    const float* Wr0 = W + (size_t)(n0 + lp) * ldw + 2 * hi;
    const float* Wr1 = Wr0 + (size_t)16 * ldw;

    v8f acc00 = {}, acc01 = {}, acc10 = {}, acc11 = {};
#pragma unroll 2
    for (int k = 0; k < K; k += 4) {
        v2f a0 = *reinterpret_cast<const v2f*>(Ar0 + k);
        v2f a1 = *reinterpret_cast<const v2f*>(Ar1 + k);
        v2f b0 = *reinterpret_cast<const v2f*>(Wr0 + k);
        v2f b1 = *reinterpret_cast<const v2f*>(Wr1 + k);
        acc00 = wmma_f32(a0, b0, acc00);
        acc01 = wmma_f32(a0, b1, acc01);
        acc10 = wmma_f32(a1, b0, acc10);
        acc11 = wmma_f32(a1, b1, acc11);
    }

    const int n = n0 + lp;
    const float bv0 = bias ? bias[n] : 0.0f;
    const float bv1 = bias ? bias[n + 16] : 0.0f;
#pragma unroll
    for (int i = 0; i < 8; ++i) {
        const int r0 = m0 + 8 * hi + i;      // C layout: vgpr i -> row i (+8 hi)
        const int r1 = r0 + 16;
        C[(size_t)r0 * ldc + n]      = acc00[i] + bv0;
        C[(size_t)r0 * ldc + n + 16] = acc01[i] + bv1;
        C[(size_t)r1 * ldc + n]      = acc10[i] + bv0;
        C[(size_t)r1 * ldc + n + 16] = acc11[i] + bv1;
    }
}

// ---------------------------------------------------------------------------
// Fused attention scores: rows r = l*B+b of enc viewed as [L*B, H].
// score[r] = sum_j tanh( (enc_r . We_j) + hW[b,j] + bias[j] ) * v[j]
// One block per 16 rows; wave w keeps 8 strip-accumulators register-resident
// (strips j0=(w+8s)*16) so each A fragment is loaded from global exactly once
// per K-step. Deterministic: half-wave shuffle reduce + LDS cross-wave reduce.
// ---------------------------------------------------------------------------
__global__ __launch_bounds__(256) void attn_score_kernel(
    const float* __restrict__ Ein,   // [L*B, H]
    const float* __restrict__ We,    // row j at We + j*2H (already offset by +H)
    const float* __restrict__ hW,    // [B, H]
    const float* __restrict__ bias,  // [H]
    const float* __restrict__ vvec,  // [H]
    float* __restrict__ scores)      // [L*B]
{
    __shared__ float part[8][16];
    const int lane = threadIdx.x & 31;
    const int wave = threadIdx.x >> 5;
    const int hi   = lane >> 4;
    const int lp   = lane & 15;
    const int m0 = blockIdx.x * 16;

    const float* Ar = Ein + (size_t)(m0 + lp) * H_ + 2 * hi;

    const float* Wr[8];
#pragma unroll
    for (int s = 0; s < 8; ++s)
        Wr[s] = We + (size_t)((wave + 8 * s) * 16 + lp) * (2 * H_) + 2 * hi;

    v8f acc[8];
#pragma unroll
    for (int s = 0; s < 8; ++s) acc[s] = (v8f){};

    for (int k = 0; k < H_; k += 4) {
        v2f a = *reinterpret_cast<const v2f*>(Ar + k);
#pragma unroll
        for (int s = 0; s < 8; ++s) {
            v2f b = *reinterpret_cast<const v2f*>(Wr[s] + k);
            acc[s] = wmma_f32(a, b, acc[s]);
        }
    }

    float rsum[8];
#pragma unroll
    for (int i = 0; i < 8; ++i) rsum[i] = 0.0f;

#pragma unroll
    for (int s = 0; s < 8; ++s) {
        const int j = (wave + 8 * s) * 16 + lp;
        const float bj = bias[j];
        const float vj = vvec[j];
        float contrib[8];
#pragma unroll
        for (int i = 0; i < 8; ++i) {
            const int r = m0 + 8 * hi + i;
            const int b = r & (B_ - 1);            // r = l*B + b, B=256
            const float tv = tanhf(acc[s][i] + hW[(size_t)b * H_ + j] + bj);
            contrib[i] = tv * vj;
        }
        // reduce the 16 columns held across the 16 lanes of this half-wave
#pragma unroll
        for (int off = 1; off < 16; off <<= 1) {
#pragma unroll
            for (int i = 0; i < 8; ++i)
                contrib[i] += __shfl_xor(contrib[i], off, 32);
        }
#pragma unroll
        for (int i = 0; i < 8; ++i) rsum[i] += contrib[i];
    }

    if (lp == 0) {
#pragma unroll
        for (int i = 0; i < 8; ++i) part[wave][8 * hi + i] = rsum[i];
    }
    __syncthreads();
    if (threadIdx.x < 16) {
        float s = 0.0f;
        for (int w = 0; w < 8; ++w) s += part[w][threadIdx.x];
        scores[m0 + threadIdx.x] = s;
    }
}

// softmax over l (stride B layout, in-place). 256 threads = one per batch row.
__global__ __launch_bounds__(256) void softmax_L_kernel(float* __restrict__ s)
{
    const int b = threadIdx.x;
    float m = -1e30f;
    for (int l = 0; l < L_; ++l) m = fmaxf(m, s[l * B_ + b]);
    float sum = 0.0f;
    for (int l = 0; l < L_; ++l) sum += expf(s[l * B_ + b] - m);
    const float inv = 1.0f / sum;
    for (int l = 0; l < L_; ++l) s[l * B_ + b] = expf(s[l * B_ + b] - m) * inv;
}

// ctx[b,h] = sum_l a[l,b] * enc[l,b,h]; written to two destinations.
__global__ __launch_bounds__(256) void ctx_kernel(
    const float* __restrict__ enc, const float* __restrict__ aw,
    float* __restrict__ o1, int ld1, float* __restrict__ o2, int ld2)
{
    const int b = blockIdx.x;
    const int h = blockIdx.y * 256 + threadIdx.x;
    float s = 0.0f;
    for (int l = 0; l < L_; ++l)
        s += aw[l * B_ + b] * enc[((size_t)l * B_ + b) * H_ + h];
    o1[(size_t)b * ld1 + h] = s;
    o2[(size_t)b * ld2 + h] = s;
}

// embedding gather into gru_in[:, 0:E]
__global__ __launch_bounds__(256) void embed_kernel(
    const float* __restrict__ emb, const int* __restrict__ idx,
    float* __restrict__ gru_in)
{
    const int b = blockIdx.x;
    const int e = blockIdx.y * 256 + threadIdx.x;
    gru_in[(size_t)b * (E_ + 2 * H_) + e] = emb[(size_t)idx[b] * E_ + e];
}

// GRU gates (PyTorch order r,z,n); writes h1 into gen_in[:,2H:3H] and outputs.
__global__ __launch_bounds__(256) void gru_gate_kernel(
    const float* __restrict__ gi, const float* __restrict__ gh,
    const float* __restrict__ b_ih, const float* __restrict__ b_hh,
    const float* __restrict__ h0,
    float* __restrict__ gen_in, float* __restrict__ h1a, float* __restrict__ h1b)
{
    const int b = blockIdx.x;
    const int j = blockIdx.y * 256 + threadIdx.x;
    const size_t o = (size_t)b * 3 * H_;
    const float ir  = gi[o + j]          + b_ih[j];
    const float iz  = gi[o + H_ + j]     + b_ih[H_ + j];
    const float inn = gi[o + 2 * H_ + j] + b_ih[2 * H_ + j];
    const float hr  = gh[o + j]          + b_hh[j];
    const float hz  = gh[o + H_ + j]     + b_hh[H_ + j];
    const float hn  = gh[o + 2 * H_ + j] + b_hh[2 * H_ + j];
    const float r = 1.0f / (1.0f + expf(-(ir + hr)));
    const float z = 1.0f / (1.0f + expf(-(iz + hz)));
    const float n = tanhf(inn + r * hn);
    const float h1 = (1.0f - z) * n + z * h0[(size_t)b * H_ + j];
    gen_in[o + 2 * H_ + j] = h1;
    h1a[(size_t)b * H_ + j] = h1;
    h1b[(size_t)b * H_ + j] = h1;
}

// row softmax over V=32000: online max/sum pass + write pass (2 passes).
__global__ __launch_bounds__(256) void softmax_V_kernel(
    const float* __restrict__ x, float* __restrict__ y)
{
    __shared__ float mred[256];
    __shared__ float sred[256];
    const int b = blockIdx.x;
    const int tid = threadIdx.x;
    const float* row = x + (size_t)b * V_;

    float m = -1e30f, s = 0.0f;
    for (int i = tid; i < V_; i += 256) {
        const float v = row[i];
        if (v > m) { s *= expf(m - v); m = v; }
        s += expf(v - m);
    }
    mred[tid] = m; sred[tid] = s;
    __syncthreads();
    for (int off = 128; off > 0; off >>= 1) {
        if (tid < off) {
            const float m2 = mred[tid + off], s2 = sred[tid + off];
            const float M = fmaxf(mred[tid], m2);
            sred[tid] = sred[tid] * expf(mred[tid] - M) + s2 * expf(m2 - M);
            mred[tid] = M;
        }
        __syncthreads();
    }
    const float M = mred[0];
    const float inv = 1.0f / sred[0];
    float* orow = y + (size_t)b * V_;
    for (int i = tid; i < V_; i += 256) orow[i] = expf(row[i] - M) * inv;
}

extern "C" void kernel_launch(void* const* d_in, const int* in_sizes, int n_in,
                              void* d_out, int out_size, void* d_ws, size_t ws_size,
                              hipStream_t stream)
{
    (void)in_sizes; (void)n_in; (void)out_size; (void)ws_size;

    const float* slot_enc = (const float*)d_in[0];
    const float* pers_enc = (const float*)d_in[2];
    const int*   m_t      = (const int*)  d_in[4];
    const float* h0       = (const float*)d_in[5];   // [1,B,H] -> [B,H]
    const float* emb      = (const float*)d_in[6];
    const float* aW       = (const float*)d_in[7];   // [H, 2H]
    const float* ab       = (const float*)d_in[8];
    const float* av       = (const float*)d_in[9];
    const float* pW       = (const float*)d_in[10];
    const float* pb       = (const float*)d_in[11];
    const float* pv       = (const float*)d_in[12];
    const float* W_ih     = (const float*)d_in[13];  // [3H, E+2H]
    const float* W_hh     = (const float*)d_in[14];  // [3H, H]
    const float* b_ih     = (const float*)d_in[15];
    const float* b_hh     = (const float*)d_in[16];
    const float* projW    = (const float*)d_in[17];  // [V, 3H]
    const float* projb    = (const float*)d_in[18];

    // workspace layout (floats)
    float* ws     = (float*)d_ws;
    float* hWa    = ws;                                 // B*H
    float* hWp    = hWa + (size_t)B_ * H_;              // B*H
    float* sca    = hWp + (size_t)B_ * H_;              // L*B
    float* scp    = sca + (size_t)L_ * B_;              // L*B
    float* gru_in = scp + (size_t)L_ * B_;              // B*(E+2H)
    float* gen_in = gru_in + (size_t)B_ * (E_ + 2 * H_);// B*3H
    float* gi     = gen_in + (size_t)B_ * 3 * H_;       // B*3H
    float* gh     = gi + (size_t)B_ * 3 * H_;           // B*3H
    float* logits = gh + (size_t)B_ * 3 * H_;           // B*V

    float* out     = (float*)d_out;                     // proba [B,V]
    float* out_h1a = out + (size_t)B_ * V_;             // h1 [1,B,H]
    float* out_h1b = out_h1a + (size_t)B_ * H_;         // h1 [1,B,H]

    // 1) embedding gather into gru_in[:, :E]
    embed_kernel<<<dim3(B_, E_ / 256), 256, 0, stream>>>(emb, m_t, gru_in);

    // 2) hW = h0 @ W_h^T for both attentions (W_h = W[:, :H], row stride 2H)
    gemm_xwT<<<dim3(H_ / 256, B_ / 32), 256, 0, stream>>>(
        h0, H_, aW, 2 * H_, nullptr, hWa, H_, H_);
    gemm_xwT<<<dim3(H_ / 256, B_ / 32), 256, 0, stream>>>(
        h0, H_, pW, 2 * H_, nullptr, hWp, H_, H_);

    // 3) fused energy GEMM + tanh + dot(v): scores [L,B]
    attn_score_kernel<<<dim3((L_ * B_) / 16), 256, 0, stream>>>(
        slot_enc, aW + H_, hWa, ab, av, sca);
    attn_score_kernel<<<dim3((L_ * B_) / 16), 256, 0, stream>>>(
        pers_enc, pW + H_, hWp, pb, pv, scp);

    // 4) softmax over L (in place)
    softmax_L_kernel<<<1, B_, 0, stream>>>(sca);
    softmax_L_kernel<<<1, B_, 0, stream>>>(scp);

    // 5) contexts -> gru_in[:, E:E+H], [E+H:] and gen_in[:, 0:H], [H:2H]
    ctx_kernel<<<dim3(B_, H_ / 256), 256, 0, stream>>>(
        slot_enc, sca, gru_in + E_, E_ + 2 * H_, gen_in, 3 * H_);
    ctx_kernel<<<dim3(B_, H_ / 256), 256, 0, stream>>>(
        pers_enc, scp, gru_in + E_ + H_, E_ + 2 * H_, gen_in + H_, 3 * H_);

    // 6) GRU input/hidden GEMMs
    gemm_xwT<<<dim3(3 * H_ / 256, B_ / 32), 256, 0, stream>>>(
        gru_in, E_ + 2 * H_, W_ih, E_ + 2 * H_, nullptr, gi, 3 * H_, E_ + 2 * H_);
    gemm_xwT<<<dim3(3 * H_ / 256, B_ / 32), 256, 0, stream>>>(
        h0, H_, W_hh, H_, nullptr, gh, 3 * H_, H_);

    // 7) gates -> h1 (gen_in[:,2H:], two output copies)
    gru_gate_kernel<<<dim3(B_, H_ / 256), 256, 0, stream>>>(
        gi, gh, b_ih, b_hh, h0, gen_in, out_h1a, out_h1b);

    // 8) projection: logits = gen_in @ projW^T + projb (393 MB stream: the
    //    memory-bound heart of the kernel, ~17 us at 23.3 TB/s)
    gemm_xwT<<<dim3(V_ / 256, B_ / 32), 256, 0, stream>>>(
        gen_in, 3 * H_, projW, 3 * H_, projb, logits, V_, 3 * H_);

    // 9) softmax over V -> proba
    softmax_V_kernel<<<B_, 256, 0, stream>>>(logits, out);
}